// Model_27401891349004
// MI455X (gfx1250) — compile-verified
//
#include <hip/hip_runtime.h>
#include <hip/hip_bf16.h>
#include <cstdint>

// ---------------------------------------------------------------------------
// GGNN link prediction for MI455X (gfx1250, wave32).
// - All GEMMs via V_WMMA_F32_16X16X4_F32 (full f32 precision, matches ref).
// - segment_sum via global f32 atomics (memory-bound term: ~307MB/timestep).
// - GRU stages activation tiles into LDS with GLOBAL_LOAD_ASYNC_TO_LDS_B128
//   (ASYNCcnt-tracked, bypasses VGPRs) + s_wait_asynccnt.
// - k-loops fully unrolled to kill accumulator-rotation mov churn.
// ---------------------------------------------------------------------------

typedef float v2f __attribute__((ext_vector_type(2)));
typedef float v8f __attribute__((ext_vector_type(8)));

__device__ __forceinline__ v8f wmma4(v2f a, v2f b, v8f c) {
  // D = A(16x4 f32) * B(4x16 f32) + C(16x16 f32)
  return __builtin_amdgcn_wmma_f32_16x16x4_f32(false, a, false, b, (short)0, c,
                                               false, false);
}

__device__ __forceinline__ float sigmoidf_fast(float x) {
  return 1.0f / (1.0f + __expf(-x));
}
__device__ __forceinline__ float tanhf_fast(float x) {
  return 2.0f * sigmoidf_fast(2.0f * x) - 1.0f;
}

// Accumulate X(16x64, row stride 64) @ W(64xLDB row-major)[:, cb:cb+64] into acc[4].
// A-frag (ISA 7.12.2, 32-bit A 16x4): lane lo holds M=lo, VGPR0 K=k0+2*hi, VGPR1 K+1.
// B-frag symmetric: lane lo holds N=lo, VGPR0 K=k0+2*hi, VGPR1 K+1.
template <int LDB>
__device__ __forceinline__ void gemm_acc(const float* __restrict__ X,
                                         const float* __restrict__ W, int cb,
                                         int lo, int hi, v8f acc[4]) {
#pragma unroll
  for (int kk = 0; kk < 16; ++kk) {
    const int kA = kk * 4 + hi * 2;
    v2f a;
    a.x = X[lo * 64 + kA];
    a.y = X[lo * 64 + kA + 1];
#pragma unroll
    for (int c = 0; c < 4; ++c) {
      const int col = cb + c * 16 + lo;
      v2f b;
      b.x = W[kA * LDB + col];
      b.y = W[(kA + 1) * LDB + col];
      acc[c] = wmma4(a, b, acc[c]);
    }
  }
}

__device__ __forceinline__ void initbias(v8f acc[4], const float* __restrict__ b1,
                                         const float* __restrict__ b2, int cb,
                                         int lo) {
#pragma unroll
  for (int c = 0; c < 4; ++c) {
    float bv = b1[cb + c * 16 + lo];
    if (b2) bv += b2[cb + c * 16 + lo];
#pragma unroll
    for (int j = 0; j < 8; ++j) acc[c][j] = bv;
  }
}

// C/D layout store: element (c,j,lane) -> row j+8*hi, col c*16+lo.
__device__ __forceinline__ void store_tile(float* __restrict__ Y, size_t obase,
                                           const v8f acc[4], int lo, int hi) {
#pragma unroll
  for (int c = 0; c < 4; ++c)
#pragma unroll
    for (int j = 0; j < 8; ++j)
      Y[obase + (size_t)(j + 8 * hi) * 64 + c * 16 + lo] = acc[c][j];
}

// ---------------- ge = emb_table[emb_ind] @ Wp + bp  (N x 100 @ 100 x 64) ----
__global__ __launch_bounds__(256) void proj_kernel(
    const int* __restrict__ emb_ind, const float* __restrict__ emb,
    const float* __restrict__ Wp, const float* __restrict__ bp,
    float* __restrict__ hout, int ntiles) {
  const int wave = threadIdx.x >> 5, lane = threadIdx.x & 31;
  const int tile = blockIdx.x * 8 + wave;
  if (tile >= ntiles) return;
  __builtin_prefetch(Wp, 0, 1);  // global_prefetch_b8; weights L2-resident
  const int lo = lane & 15, hi = lane >> 4;
  const size_t erow = (size_t)emb_ind[tile * 16 + lo] * 100;
  v8f acc[4];
  initbias(acc, bp, nullptr, 0, lo);
#pragma unroll
  for (int kk = 0; kk < 25; ++kk) {  // K = 100
    const int kA = kk * 4 + hi * 2;
    v2f a;
    a.x = emb[erow + kA];
    a.y = emb[erow + kA + 1];
#pragma unroll
    for (int c = 0; c < 4; ++c) {
      const int col = c * 16 + lo;
      v2f b;
      b.x = Wp[(size_t)kA * 64 + col];
      b.y = Wp[(size_t)(kA + 1) * 64 + col];
      acc[c] = wmma4(a, b, acc[c]);
    }
  }
  store_tile(hout, (size_t)tile * 1024, acc, lo, hi);
}

// ---------------- m = h @ Wm + bm  (N x 64 @ 64 x 64) -----------------------
__global__ __launch_bounds__(256) void msg_kernel(const float* __restrict__ h,
                                                  const float* __restrict__ Wm,
                                                  const float* __restrict__ bm,
                                                  float* __restrict__ m,
                                                  int ntiles) {
  const int wave = threadIdx.x >> 5, lane = threadIdx.x & 31;
  const int tile = blockIdx.x * 8 + wave;
  if (tile >= ntiles) return;
  __builtin_prefetch(Wm, 0, 1);
  const int lo = lane & 15, hi = lane >> 4;
  v8f acc[4];
  initbias(acc, bm, nullptr, 0, lo);
  gemm_acc<64>(h + (size_t)tile * 1024, Wm, 0, lo, hi, acc);
  store_tile(m, (size_t)tile * 1024, acc, lo, hi);
}

// ---------------- agg[tgt] += m[src] over edges -----------------------------
__global__ __launch_bounds__(256) void scatter_kernel(
    const int* __restrict__ adj, const float* __restrict__ m,
    float* __restrict__ agg, int E) {
  const int gid = blockIdx.x * blockDim.x + threadIdx.x;
  if (gid >= E * 4) return;
  const int e = gid >> 2;
  const int fo = (gid & 3) * 16;
  const int src = adj[2 * e + 0];
  const int tgt = adj[2 * e + 1];
  const float4* ms = (const float4*)(m + (size_t)src * 64 + fo);
  float* ag = agg + (size_t)tgt * 64 + fo;
#pragma unroll
  for (int q = 0; q < 4; ++q) {
    float4 v = ms[q];
    atomicAdd(ag + 4 * q + 0, v.x);
    atomicAdd(ag + 4 * q + 1, v.y);
    atomicAdd(ag + 4 * q + 2, v.z);
    atomicAdd(ag + 4 * q + 3, v.w);
  }
}

// ---------------- fused GRU cell: h' = GRU(agg, h) --------------------------
__global__ __launch_bounds__(256) void gru_kernel(
    const float* __restrict__ agg, const float* __restrict__ h,
    const float* __restrict__ Wih, const float* __restrict__ Whh,
    const float* __restrict__ bih, const float* __restrict__ bhh,
    float* __restrict__ hnew, int ntiles) {
  __shared__ float lds[8 * 2048];  // per wave: 16x64 agg tile + 16x64 h tile
  const int wave = threadIdx.x >> 5, lane = threadIdx.x & 31;
  const int tile = blockIdx.x * 8 + wave;
  if (tile >= ntiles) return;
  __builtin_prefetch(Wih, 0, 1);
  __builtin_prefetch(Whh, 0, 1);
  const int lo = lane & 15, hi = lane >> 4;
  float* aT = lds + wave * 2048;
  float* hT = aT + 1024;
  const size_t base = (size_t)tile * 1024;

  // Async global->LDS staging (CDNA5 path, ASYNCcnt-tracked, bypasses VGPRs).
  // LDS VDST operand = per-lane LDS byte address = low 32 bits of generic ptr
  // (LDS aperture: LDS address is the low 32 bits).
  {
    const unsigned ldsA = (unsigned)(uintptr_t)aT;
    const unsigned ldsH = (unsigned)(uintptr_t)hT;
    const float* ga = agg + base;
    const float* gb = h + base;
#pragma unroll
    for (int i = 0; i < 8; ++i) {
      const unsigned el = (unsigned)(i * 32 + lane) * 16u;  // 16B per lane
      asm volatile("global_load_async_to_lds_b128 %0, %1, off"
                   :
                   : "v"(ldsA + el), "v"(ga + (i * 32 + lane) * 4)
                   : "memory");
      asm volatile("global_load_async_to_lds_b128 %0, %1, off"
                   :
                   : "v"(ldsH + el), "v"(gb + (i * 32 + lane) * 4)
                   : "memory");
    }
    asm volatile("s_wait_asynccnt 0x0" ::: "memory");
  }

  v8f r[4], z[4], gi[4], gh[4];
  // r = sigmoid(agg@Wih[:,0:64] + h@Whh[:,0:64] + bih + bhh)
  initbias(r, bih, bhh, 0, lo);
  gemm_acc<192>(aT, Wih, 0, lo, hi, r);
  gemm_acc<192>(hT, Whh, 0, lo, hi, r);
  // z = sigmoid(... cols 64:128)
  initbias(z, bih, bhh, 64, lo);
  gemm_acc<192>(aT, Wih, 64, lo, hi, z);
  gemm_acc<192>(hT, Whh, 64, lo, hi, z);
  // ig, hg (cols 128:192, kept separate for n = tanh(ig + r*hg))
  initbias(gi, bih, nullptr, 128, lo);
  gemm_acc<192>(aT, Wih, 128, lo, hi, gi);
  initbias(gh, bhh, nullptr, 128, lo);
  gemm_acc<192>(hT, Whh, 128, lo, hi, gh);

#pragma unroll
  for (int c = 0; c < 4; ++c)
#pragma unroll
    for (int j = 0; j < 8; ++j) {
      const float rv = sigmoidf_fast(r[c][j]);
      const float zv = sigmoidf_fast(z[c][j]);
      const float nv = tanhf_fast(gi[c][j] + rv * gh[c][j]);
      const float hv = hT[(j + 8 * hi) * 64 + c * 16 + lo];
      hnew[base + (size_t)(j + 8 * hi) * 64 + c * 16 + lo] =
          (1.0f - zv) * nv + zv * hv;
    }
}

// ---------------- gather rows for readout -----------------------------------
__global__ void gather_kernel(const float* __restrict__ h,
                              const int* __restrict__ ind,
                              float* __restrict__ outp, int B) {
  const int gid = blockIdx.x * blockDim.x + threadIdx.x;
  if (gid >= B * 64) return;
  const int b = gid >> 6, f = gid & 63;
  outp[gid] = h[(size_t)ind[b] * 64 + f];
}

__global__ void zero_kernel(float* __restrict__ p, int n) {
  const int gid = blockIdx.x * blockDim.x + threadIdx.x;
  if (gid < n) p[gid] = 0.0f;
}

__global__ void init_loss_kernel(float* __restrict__ out, int B) {
  if (threadIdx.x == 0 && blockIdx.x == 0) out[B] = 0.0f;
}

// ---------------- readout: concat -> W1/relu -> W2 -> sigmoid + BCE ---------
__global__ __launch_bounds__(256) void readout_kernel(
    const float* __restrict__ p0, const float* __restrict__ p1,
    const float* __restrict__ W1, const float* __restrict__ b1,
    const float* __restrict__ W2, const float* __restrict__ b2,
    const int* __restrict__ labels, float* __restrict__ out, int ntiles,
    int B) {
  const int wave = threadIdx.x >> 5, lane = threadIdx.x & 31;
  const int tile = blockIdx.x * 8 + wave;
  if (tile >= ntiles) return;
  __builtin_prefetch(W1, 0, 1);
  const int lo = lane & 15, hi = lane >> 4;
  v8f acc[4];
  initbias(acc, b1, nullptr, 0, lo);
#pragma unroll
  for (int kk = 0; kk < 32; ++kk) {  // K = 128 (concat of two 64-dim halves)
    const int kA = kk * 4 + hi * 2;
    const float* src = (kk < 16) ? p0 : p1;
    const int kloc = (kk < 16) ? kA : kA - 64;
    v2f a;
    a.x = src[(size_t)(tile * 16 + lo) * 64 + kloc];
    a.y = src[(size_t)(tile * 16 + lo) * 64 + kloc + 1];
#pragma unroll
    for (int c = 0; c < 4; ++c) {
      const int col = c * 16 + lo;
      v2f b;
      b.x = W1[(size_t)kA * 64 + col];
      b.y = W1[(size_t)(kA + 1) * 64 + col];
      acc[c] = wmma4(a, b, acc[c]);
    }
  }
  // relu(hidden) . W2  (per-row dot, cross-lane reduce within each 16-lane half)
  float w2l[4];
#pragma unroll
  for (int c = 0; c < 4; ++c) w2l[c] = W2[c * 16 + lo];
  float zp[8];
#pragma unroll
  for (int j = 0; j < 8; ++j) {
    float s = 0.0f;
#pragma unroll
    for (int c = 0; c < 4; ++c) s += fmaxf(acc[c][j], 0.0f) * w2l[c];
    zp[j] = s;
  }
  for (int off = 1; off < 16; off <<= 1) {
#pragma unroll
    for (int j = 0; j < 8; ++j) zp[j] += __shfl_xor(zp[j], off, 32);
  }
  if (lo == 0) {
    const float bb = b2[0];
    const float invB = 1.0f / (float)B;
    for (int j = 0; j < 8; ++j) {
      const int row = tile * 16 + j + 8 * hi;
      const float zz = zp[j] + bb;
      out[row] = sigmoidf_fast(zz);  // probs
      const float y = (float)labels[row];
      // stable log_sigmoid
      const float lsp = fminf(zz, 0.0f) - __logf(1.0f + __expf(-fabsf(zz)));
      const float lsn = fminf(-zz, 0.0f) - __logf(1.0f + __expf(-fabsf(zz)));
      const float term = y * lsp + (1.0f - y) * lsn;
      atomicAdd(out + B, -term * invB);
    }
  }
}

// ---------------------------------------------------------------------------
extern "C" void kernel_launch(void* const* d_in, const int* in_sizes, int n_in,
                              void* d_out, int out_size, void* d_ws,
                              size_t ws_size, hipStream_t stream) {
  (void)n_in;
  (void)out_size;
  (void)ws_size;
  const int N = in_sizes[0];
  const int E = in_sizes[2] / 2;
  const int B = in_sizes[4];

  const int* emb_ind[2] = {(const int*)d_in[0], (const int*)d_in[1]};
  const int* adj[2] = {(const int*)d_in[2], (const int*)d_in[3]};
  const int* prop[2] = {(const int*)d_in[4], (const int*)d_in[5]};
  const int* labels = (const int*)d_in[6];
  const float* emb_table = (const float*)d_in[7];
  const float* Wp = (const float*)d_in[8];
  const float* bp = (const float*)d_in[9];
  const float* Wm[2] = {(const float*)d_in[10], (const float*)d_in[16]};
  const float* bm[2] = {(const float*)d_in[11], (const float*)d_in[17]};
  const float* Wih[2] = {(const float*)d_in[12], (const float*)d_in[18]};
  const float* Whh[2] = {(const float*)d_in[13], (const float*)d_in[19]};
  const float* bih[2] = {(const float*)d_in[14], (const float*)d_in[20]};
  const float* bhh[2] = {(const float*)d_in[15], (const float*)d_in[21]};
  const float* W1 = (const float*)d_in[22];
  const float* b1 = (const float*)d_in[23];
  const float* W2 = (const float*)d_in[24];
  const float* b2 = (const float*)d_in[25];
  float* out = (float*)d_out;

  float* ws = (float*)d_ws;
  float* bufA = ws;                            // N*64
  float* bufB = bufA + (size_t)N * 64;         // N*64
  float* aggBuf = bufB + (size_t)N * 64;       // N*64
  float* prop_out[2];
  prop_out[0] = aggBuf + (size_t)N * 64;       // B*64
  prop_out[1] = prop_out[0] + (size_t)B * 64;  // B*64

  const int ntiles = N / 16;  // N divisible by 16
  const dim3 blk(256);
  const int gemmBlocks = (ntiles + 7) / 8;
  const int zeroBlocks = (N * 64 + 255) / 256;
  const int scatBlocks = (E * 4 + 255) / 256;

  for (int g = 0; g < 2; ++g) {
    float* h = bufA;
    float* alt = bufB;
    proj_kernel<<<gemmBlocks, blk, 0, stream>>>(emb_ind[g], emb_table, Wp, bp,
                                                h, ntiles);
    for (int l = 0; l < 2; ++l) {
      for (int t = 0; t < 3; ++t) {
        msg_kernel<<<gemmBlocks, blk, 0, stream>>>(h, Wm[l], bm[l], alt,
                                                   ntiles);
        zero_kernel<<<zeroBlocks, blk, 0, stream>>>(aggBuf, N * 64);
        scatter_kernel<<<scatBlocks, blk, 0, stream>>>(adj[g], alt, aggBuf, E);
        gru_kernel<<<gemmBlocks, blk, 0, stream>>>(aggBuf, h, Wih[l], Whh[l],
                                                   bih[l], bhh[l], alt, ntiles);
        float* tmp = h;
        h = alt;
        alt = tmp;
      }
    }
    gather_kernel<<<(B * 64 + 255) / 256, blk, 0, stream>>>(h, prop[g],
                                                            prop_out[g], B);
  }
  init_loss_kernel<<<1, 32, 0, stream>>>(out, B);
  readout_kernel<<<(B / 16 + 7) / 8, blk, 0, stream>>>(
      prop_out[0], prop_out[1], W1, b1, W2, b2, labels, out, B / 16, B);
}